// Quantize_86749749445197
// MI455X (gfx1250) — compile-verified
//
#include <hip/hip_runtime.h>
#include <hip/hip_bf16.h>

typedef float v2f __attribute__((ext_vector_type(2)));
typedef float v8f __attribute__((ext_vector_type(8)));

#define DD 64
#define KK 1024

// Kernel 1: e2[k] = sum_d embed[d][k]^2   (embed is [64][1024] row-major)
__global__ void vq_e2_kernel(const float* __restrict__ embed, float* __restrict__ e2) {
    int k = blockIdx.x * blockDim.x + threadIdx.x;
    if (k >= KK) return;
    float s = 0.0f;
    #pragma unroll
    for (int d = 0; d < DD; ++d) {
        float v = embed[d * KK + k];
        s += v * v;
    }
    e2[k] = s;
}

// Kernel 2: per wave: 4 row-tiles of 16 tokens; loop K in 16-code tiles.
// score = e2[k] - 2*x.e: WMMA f32 16x16x4 with A pre-scaled by -2, C init 0,
// e2 added at compare time from LDS (DScnt path — doesn't drain the global
// B-prefetch pipeline). B fragments register double-buffered across tiles.
__global__ __launch_bounds__(128) void vq_main_kernel(
    const float* __restrict__ x,       // [N][64]
    const float* __restrict__ embed,   // [64][1024]
    const float* __restrict__ e2g,     // [1024]
    float* __restrict__ outQ,          // [N][64]
    float* __restrict__ outDiff,       // [N][64]
    float* __restrict__ outInd)        // [N] (indices as float)
{
    __shared__ float se2[KK];          // 4 KB: e2 staged in LDS
    __shared__ int sidx[4][64];

    const int lane = threadIdx.x & 31;
    const int wave = threadIdx.x >> 5;
    const int half = lane >> 4;     // 0: lanes 0-15, 1: lanes 16-31
    const int m    = lane & 15;
    const int rowBase = blockIdx.x * 256 + wave * 64;

    // ---- Stage e2 into LDS (block-cooperative) ----
    #pragma unroll
    for (int i = 0; i < KK / 128; ++i)
        se2[i * 128 + threadIdx.x] = e2g[i * 128 + threadIdx.x];

    // ---- Load A fragments: a[t][s] holds X[row][4s+2h .. 4s+2h+1] * (-2) ----
    v2f a[4][16];
    #pragma unroll
    for (int t = 0; t < 4; ++t) {
        const float* xr = x + (size_t)(rowBase + t * 16 + m) * DD;
        #pragma unroll
        for (int s = 0; s < 16; ++s) {
            int kd = 4 * s + 2 * half;
            v2f v = *(const v2f*)(xr + kd);
            a[t][s].x = -2.0f * v.x;
            a[t][s].y = -2.0f * v.y;
        }
    }

    float best[4][8];
    int   bidx[4][8];
    #pragma unroll
    for (int t = 0; t < 4; ++t)
        #pragma unroll
        for (int j = 0; j < 8; ++j) { best[t][j] = 3.4e38f; bidx[t][j] = 0; }

    __syncthreads();   // se2 ready

    // ---- Prologue: prefetch B fragments for K-tile 0 ----
    v2f bnext[16];
    {
        const float* ecol = embed + m + 2 * half * KK;
        #pragma unroll
        for (int s = 0; s < 16; ++s) {
            bnext[s].x = ecol[(4 * s) * KK];
            bnext[s].y = ecol[(4 * s + 1) * KK];
        }
    }

    const v8f czero = {0.f, 0.f, 0.f, 0.f, 0.f, 0.f, 0.f, 0.f};

    // ---- K loop: 64 tiles of 16 codes, double-buffered B ----
    #pragma unroll 2
    for (int kt = 0; kt < 64; ++kt) {
        const int kk = kt * 16 + m;           // this lane's code column

        // consume current buffer
        v2f bcur[16];
        #pragma unroll
        for (int s = 0; s < 16; ++s) bcur[s] = bnext[s];

        // issue prefetch for next tile (wraps on last trip; harmless)
        {
            const int ktn = (kt + 1) & 63;
            const float* ecol = embed + (ktn * 16 + m) + 2 * half * KK;
            #pragma unroll
            for (int s = 0; s < 16; ++s) {
                bnext[s].x = ecol[(4 * s) * KK];
                bnext[s].y = ecol[(4 * s + 1) * KK];
            }
        }

        const float ev = se2[kk];             // DS load: cheap, separate counter

        v8f c[4] = {czero, czero, czero, czero};

        #pragma unroll
        for (int s = 0; s < 16; ++s) {
            #pragma unroll
            for (int t = 0; t < 4; ++t) {
                c[t] = __builtin_amdgcn_wmma_f32_16x16x4_f32(
                    false, a[t][s], false, bcur[s], (short)0, c[t], false, false);
            }
        }

        #pragma unroll
        for (int t = 0; t < 4; ++t)
            #pragma unroll
            for (int j = 0; j < 8; ++j) {
                float sc = c[t][j] + ev;
                if (sc < best[t][j]) { best[t][j] = sc; bidx[t][j] = kk; }
            }
    }

    // ---- Butterfly argmin within each 16-lane half (prefer smaller idx on tie) ----
    #pragma unroll
    for (int off = 1; off < 16; off <<= 1) {
        #pragma unroll
        for (int t = 0; t < 4; ++t)
            #pragma unroll
            for (int j = 0; j < 8; ++j) {
                float ov = __shfl_xor(best[t][j], off, 32);
                int   oi = __shfl_xor(bidx[t][j], off, 32);
                if (ov < best[t][j] || (ov == best[t][j] && oi < bidx[t][j])) {
                    best[t][j] = ov; bidx[t][j] = oi;
                }
            }
    }

    // lane 0 holds rows j (M=j), lane 16 holds rows j+8 of each tile
    if (m == 0) {
        #pragma unroll
        for (int t = 0; t < 4; ++t)
            #pragma unroll
            for (int j = 0; j < 8; ++j)
                sidx[wave][t * 16 + half * 8 + j] = bidx[t][j];
    }
    __syncthreads();

    // ---- Write indices (as float) ----
    outInd[rowBase + lane]      = (float)sidx[wave][lane];
    outInd[rowBase + lane + 32] = (float)sidx[wave][lane + 32];

    // ---- Gather + quantize/diff: lane covers 2 columns of each of 64 rows ----
    const int c0 = lane * 2;
    for (int rr = 0; rr < 64; ++rr) {
        const int kq  = sidx[wave][rr];
        const int row = rowBase + rr;
        float q0 = embed[(size_t)c0 * KK + kq];
        float q1 = embed[(size_t)(c0 + 1) * KK + kq];
        v2f xv = *(const v2f*)(x + (size_t)row * DD + c0);
        v2f qv; qv.x = q0; qv.y = q1;
        v2f dv;
        dv.x = (q0 - xv.x) * (q0 - xv.x);
        dv.y = (q1 - xv.y) * (q1 - xv.y);
        *(v2f*)(outQ    + (size_t)row * DD + c0) = qv;
        *(v2f*)(outDiff + (size_t)row * DD + c0) = dv;
    }
}

extern "C" void kernel_launch(void* const* d_in, const int* in_sizes, int n_in,
                              void* d_out, int out_size, void* d_ws, size_t ws_size,
                              hipStream_t stream) {
    const float* x     = (const float*)d_in[0];   // [16,4096,64] f32
    const float* embed = (const float*)d_in[1];   // [64,1024]    f32
    const int N = in_sizes[0] / DD;               // 65536 tokens

    float* e2 = (float*)d_ws;                     // 4 KB scratch

    float* outQ    = (float*)d_out;               // [N][64]
    float* outDiff = outQ + (size_t)N * DD;       // [N][64]
    float* outInd  = outDiff + (size_t)N * DD;    // [N]

    vq_e2_kernel<<<KK / 256, 256, 0, stream>>>(embed, e2);
    vq_main_kernel<<<N / 256, 128, 0, stream>>>(x, embed, e2, outQ, outDiff, outInd);
}